// ES_51994874085958
// MI455X (gfx1250) — compile-verified
//
#include <hip/hip_runtime.h>
#include <hip/hip_bf16.h>

// ES-RNN style exponential-smoothing forward for MI455X (gfx1250).
// Thread-per-series; y staged global->LDS with the CDNA5 Tensor Data Mover
// (one tensor_load_to_lds per 64-series x 96-step chunk, double-buffered),
// 24-entry seasonal buffer fully register-resident, float4-packed outputs.

#define N_SER   32768
#define N_TIME  512
#define SEASN   24
#define L_COLS  512          // levels
#define S_COLS  536          // 24 init + 1 repeat + 511 seasons
#define D_COLS  511          // log diffs

#define BLOCK   64
#define CHUNK_T 96           // timesteps per chunk (4 groups of 24)
#define PITCH   99           // TDM pads 1 dword per 32 -> 96+3; gcd(99,64)=1, conflict-free

typedef unsigned int v4u __attribute__((ext_vector_type(4)));
typedef int          v8i __attribute__((ext_vector_type(8)));
typedef int          v4i __attribute__((ext_vector_type(4)));

// ---- TDM: one 2D tile (64 series x 96 timesteps) global -> LDS -----------
// D# per CDNA5 ISA. Padding: pad_enable=1,
// pad_interval=4 (every 32 dwords), pad_amount=0 (1 dword) -> LDS pitch 99.
__device__ __forceinline__ void tdm_load_chunk(const float* y, int s0, int c,
                                               unsigned lds_base) {
  const int t0 = c * CHUNK_T;
  const unsigned long long ga =
      (unsigned long long)(uintptr_t)y +
      ((unsigned long long)s0 * N_TIME + (unsigned)t0) * 4ull;

  v4u g0;
  g0.x = 1u;                                             // count=1 (valid user D#)
  g0.y = lds_base;                                       // lds_addr (bytes)
  g0.z = (unsigned)(ga & 0xFFFFFFFFu);                   // global_addr[31:0]
  g0.w = (unsigned)((ga >> 32) & 0x01FFFFFFu) | (2u << 30); // addr[56:32] | type=2

  const int dim0 = N_TIME - t0;   // OOB reads return 0 -> last chunk auto-clamped
  v8i g1;
  g1[0] = (2 << 16) | (1 << 20) | (4 << 22);  // data_size=4B | pad_en | pad every 32dw by 1dw
  g1[1] = (dim0 & 0xFFFF) << 16;              // tensor_dim0[15:0]
  g1[2] = ((dim0 >> 16) & 0xFFFF) | (BLOCK << 16);  // tensor_dim0[31:16] | tensor_dim1=64
  g1[3] = (CHUNK_T << 16);                    // tensor_dim1[31:16]=0 | tile_dim0=96
  g1[4] = BLOCK;                              // tile_dim1=64 | tile_dim2=0
  g1[5] = N_TIME;                             // tensor_dim0_stride[31:0] = 512
  g1[6] = 0;                                  // stride0[47:32] | tensor_dim1_stride[15:0]
  g1[7] = 0;

  v4i z4; z4[0] = 0; z4[1] = 0; z4[2] = 0; z4[3] = 0;   // groups 2/3: 2D tensor
  v8i z8;                                                // extra arg on 6-arg toolchain
#pragma unroll
  for (int i = 0; i < 8; ++i) z8[i] = 0;
  __builtin_amdgcn_tensor_load_to_lds(g0, g1, z4, z4, z8, 0);
}

#define TENSOR_WAIT0() __builtin_amdgcn_s_wait_tensorcnt(0)

__device__ __forceinline__ void st4(float* p, const float* v) {
  float4 q; q.x = v[0]; q.y = v[1]; q.z = v[2]; q.w = v[3];
  *reinterpret_cast<float4*>(p) = q;
}

__global__ __launch_bounds__(BLOCK)
void es_forward_kernel(const float* __restrict__ y,
                       const int*   __restrict__ idxs,
                       const float* __restrict__ lev_sms_p,
                       const float* __restrict__ init_seas_p,
                       float*       __restrict__ out)
{
  __shared__ float lds_y[2][BLOCK * PITCH];

  const int tid = threadIdx.x;
  const int s0  = blockIdx.x * BLOCK;
  const int s   = s0 + tid;

  if (tid < 32)                                           // one TDM issue per block
    tdm_load_chunk(y, s0, 0, (unsigned)(uintptr_t)&lds_y[0][0]);

  // ---- per-series parameters (overlaps the TDM copy) ----------------------
  const int   idx = idxs[s];
  const float xp  = lev_sms_p[idx];
  const float a   = __fdividef(1.0f, 1.0f + __expf(-xp)); // lev_sms == seas_sms (ref quirk)
  const float b   = 1.0f - a;

  float buf[SEASN];
#pragma unroll
  for (int j = 0; j < SEASN; ++j)
    buf[j] = __expf(init_seas_p[(size_t)idx * SEASN + j]);

  float* __restrict__ lev_row  = out + (size_t)s * L_COLS;
  float* __restrict__ seas_row = out + (size_t)N_SER * L_COLS + (size_t)s * S_COLS;
  float* __restrict__ logd_row = out + (size_t)N_SER * (L_COLS + S_COLS) + (size_t)s * D_COLS;

#pragma unroll
  for (int q = 0; q < 6; ++q) st4(seas_row + 4 * q, buf + 4 * q);  // cols 0..23
  seas_row[SEASN] = buf[0];                                        // repeated col 24

  TENSOR_WAIT0();                         // chunk 0 resident (wave1: cnt already 0)
  __syncthreads();
  if (tid < 32)
    tdm_load_chunk(y, s0, 1, (unsigned)(uintptr_t)&lds_y[1][0]);   // overlap with compute

  const float* row0 = &lds_y[0][tid * PITCH];
  const float* row1 = &lds_y[1][tid * PITCH];

  float lev    = __fdividef(row0[0], buf[0]);   // lev0 = y[:,0] / init_seas[:,0]
  float loglev = __logf(lev);
  lev_row[0]   = lev;

  // One smoothing step; pos/offsets are compile-time after unrolling.
  auto step = [&](float y_t, int pos, float& olev, float& oseas, float& ologd) {
    const float seas  = buf[pos];
    const float nlev  = a * __fdividef(y_t, seas) + b * lev;
    const float nseas = a * __fdividef(y_t, nlev) + b * seas;
    const float lgl   = __logf(nlev);
    olev = nlev; oseas = nseas; ologd = lgl - loglev;
    buf[pos] = nseas;
    lev      = nlev;
    loglev   = lgl;
  };

  // 24 steps at chunk offset o0 (pos=0..23), outputs at t = tg..tg+23, packed.
  auto run_group = [&](const float* r, int o0, int tg) {
    float lv[24], sn[24], ld[24];
#pragma unroll
    for (int i = 0; i < 24; ++i) {
      const int o = o0 + i;                  // constant after unroll
      step(r[o + (o >> 5)], i, lv[i], sn[i], ld[i]);   // +1 dword pad per 32
    }
#pragma unroll
    for (int q = 0; q < 6; ++q) {
      st4(lev_row + tg + 4 * q,          lv + 4 * q);
      st4(seas_row + SEASN + tg + 4 * q, sn + 4 * q);
    }
#pragma unroll
    for (int i = 0; i < 24; ++i) logd_row[tg - 1 + i] = ld[i];
  };

  // ---- chunk 0: t = 1..95 (23-step prologue + 3 full groups) --------------
#pragma unroll
  for (int i = 1; i < 24; ++i) {             // pos = t, offsets < 32 (no pad)
    float l, sn, dd;
    step(row0[i], i, l, sn, dd);
    lev_row[i] = l; seas_row[SEASN + i] = sn; logd_row[i - 1] = dd;
  }
#pragma unroll
  for (int g = 0; g < 3; ++g) run_group(row0, 24 + g * 24, 24 + g * 24);

  // ---- chunks 1..4: prefetch c+1 while computing c ------------------------
  for (int c = 1; c <= 4; ++c) {
    TENSOR_WAIT0();                          // chunk c resident
    __syncthreads();                         // chunk c-1's buffer now reusable
    if (tid < 32)
      tdm_load_chunk(y, s0, c + 1, (unsigned)(uintptr_t)&lds_y[(c + 1) & 1][0]);
    const float* r  = (c & 1) ? row1 : row0;
    const int    t0 = c * CHUNK_T;
#pragma unroll
    for (int g = 0; g < 4; ++g) run_group(r, g * 24, t0 + g * 24);
  }

  // ---- chunk 5: t = 480..511 (one group + 8-step epilogue) ----------------
  TENSOR_WAIT0();
  __syncthreads();
  run_group(row1, 0, 480);
  {
    float lv[8], sn[8], ld[8];
#pragma unroll
    for (int i = 0; i < 8; ++i) {
      const int o = 24 + i;
      step(row1[o + (o >> 5)], i, lv[i], sn[i], ld[i]);   // t = 504+i, pos = i
    }
    st4(lev_row + 504, lv);  st4(lev_row + 508, lv + 4);
    st4(seas_row + SEASN + 504, sn);  st4(seas_row + SEASN + 508, sn + 4);
#pragma unroll
    for (int i = 0; i < 8; ++i) logd_row[503 + i] = ld[i];
  }
}

extern "C" void kernel_launch(void* const* d_in, const int* in_sizes, int n_in,
                              void* d_out, int out_size, void* d_ws, size_t ws_size,
                              hipStream_t stream) {
  (void)in_sizes; (void)n_in; (void)out_size; (void)d_ws; (void)ws_size;
  const float* y         = (const float*)d_in[0];
  const int*   idxs      = (const int*)d_in[1];
  const float* lev_sms   = (const float*)d_in[2];
  // d_in[3] (seas_sms) is dead in the reference; d_in[5]/d_in[6] are the
  // seasonality/output_size scalars, fixed at 24/7 here.
  const float* init_seas = (const float*)d_in[4];
  float*       out       = (float*)d_out;

  dim3 grid(N_SER / BLOCK), block(BLOCK);
  es_forward_kernel<<<grid, block, 0, stream>>>(y, idxs, lev_sms, init_seas, out);
}